// PPIModel_80582176407619
// MI455X (gfx1250) — compile-verified
//
#include <hip/hip_runtime.h>
#include <math.h>

#define N_NODES   100000
#define N_EDGES   3200000
#define N_HID     3
#define HEAD_OUT  1024

typedef __attribute__((ext_vector_type(2))) float v2f;
typedef __attribute__((ext_vector_type(8))) float v8f;

// ---------------- init: zero accumulators, interleave features ------------
// xv/aggv/sumv are [N] float2: component .x = feature-1 chain, .y = feature-2.
__global__ void init_kernel(const float* __restrict__ f1, const float* __restrict__ f2,
                            float* deg, v2f* xv, v2f* aggv, v2f* sumv,
                            float* o1, float* o2, int n) {
  int i = blockIdx.x * blockDim.x + threadIdx.x;
  if (i < n) {
    deg[i] = 0.f;
    v2f z; z.x = 0.f; z.y = 0.f;
    aggv[i] = z; sumv[i] = z;
    v2f x; x.x = f1[i]; x.y = f2[i];
    xv[i] = x;
  }
  if (i < HEAD_OUT) { o1[i] = 0.f; o2[i] = 0.f; }
}

// ---------------- in-degree ----------------------------------------------
__global__ void deg_kernel(const int* __restrict__ dst, float* deg, int e) {
  int i = blockIdx.x * blockDim.x + threadIdx.x;
  if (i < e) atomicAdd(&deg[dst[i]], 1.0f);
}

// ---------------- edge scatter: one b64 gather per edge, two f32 atomics --
__global__ void edge_kernel(const int* __restrict__ src, const int* __restrict__ dst,
                            const float* __restrict__ norm,
                            const v2f* __restrict__ xv, float* aggf, int e) {
  int i = blockIdx.x * blockDim.x + threadIdx.x;
  if (i < e) {
    int s = src[i], d = dst[i];
    float w = norm[i];
    v2f x = xv[s];                       // single 8B gather (one cacheline)
    atomicAdd(&aggf[2 * d + 0], x.x * w);
    atomicAdd(&aggf[2 * d + 1], x.y * w);
  }
}

// ---------------- node update: GIN linear + relu; resets agg --------------
__global__ void node_kernel(const float* __restrict__ deg,
                            v2f* xv, v2f* aggv, v2f* sumv,
                            const float* __restrict__ gw, const float* __restrict__ gb,
                            int l, int n) {
  int i = blockIdx.x * blockDim.x + threadIdx.x;
  if (i < n) {
    float idg = 1.0f / fmaxf(deg[i], 1.0f);
    float wl = gw[l], bl = gb[l];
    v2f x = xv[i], a = aggv[i], s = sumv[i];
    float r1 = (x.x + a.x * idg) * wl + bl;
    float r2 = (x.y + a.y * idg) * wl + bl;
    s.x += r1; s.y += r2;
    sumv[i] = s;
    v2f nx; nx.x = fmaxf(r1, 0.f); nx.y = fmaxf(r2, 0.f);
    xv[i] = nx;
    v2f z; z.x = 0.f; z.y = 0.f;
    aggv[i] = z;                         // fresh zeros for next layer
  }
}

// ---------------- head GEMV, both features fused, V_WMMA_F32_16X16X4_F32 --
// A (16x4 f32): row0 = feat1 sums/3, row1 = feat2 sums/3, rows 2..15 = 0.
// Branch-free lane setup; induction pointers + immediate offsets in the loop.
// W streamed once grid-wide with non-temporal hint (don't pollute L2);
// hs kept regular-temporal (re-read by all 64 column tiles, L2-resident).
#define HEAD_TILES    (HEAD_OUT / 16)   // 64 column tiles
#define HEAD_GY       16                // grid.y
#define WAVES_PER_BLK 8                 // 256-thread block = 8 wave32
#define UNROLL        8

__global__ void head_wmma_kernel(const float* __restrict__ hs,  // interleaved sums [2*N]
                                 const float* __restrict__ W,
                                 float* o1, float* o2, int n) {
  int lane = threadIdx.x & 31;
  int wave = threadIdx.x >> 5;
  int col0 = blockIdx.x * 16;
  int nChunks = gridDim.y * WAVES_PER_BLK;
  int chunk = blockIdx.y * WAVES_PER_BLK + wave;

  int KG = n >> 2;                                  // K-groups of 4
  int per = (KG + nChunks - 1) / nChunks;
  int kg0 = chunk * per;
  int kg1 = kg0 + per; if (kg1 > KG) kg1 = KG;
  int cnt = kg1 - kg0; if (cnt < 0) cnt = 0;

  int m  = lane & 15;                  // A row / B column in tile
  int kh = lane >> 4;                  // 0 -> K{0,1}, 1 -> K{2,3}
  int fsel = (m == 1) ? 1 : 0;         // which feature channel this lane carries
  float asc = (m < 2) ? (1.0f / 3.0f) : 0.0f;  // layer-mean folded in; kills rows 2..15

  // induction pointers: per-lane K-start is kg0*4 + kh*2
  int i0 = (kg0 << 2) + (kh << 1);
  const float* wp = W + (size_t)i0 * HEAD_OUT + col0 + m;
  const float* hp = hs + 2 * i0 + fsel;

  v8f c = {};

  // per-u immediate offsets: W advances 4 rows (4*HEAD_OUT) per kg, hs 8 floats
  int it = 0;
  for (; it + UNROLL <= cnt; it += UNROLL) {
    #pragma unroll
    for (int u = 0; u < UNROLL; ++u) {
      v2f a;
      a.x = hp[8 * u]     * asc;
      a.y = hp[8 * u + 2] * asc;
      v2f b;
      b.x = __builtin_nontemporal_load(wp + (size_t)u * 4 * HEAD_OUT);
      b.y = __builtin_nontemporal_load(wp + (size_t)u * 4 * HEAD_OUT + HEAD_OUT);
      c = __builtin_amdgcn_wmma_f32_16x16x4_f32(false, a, false, b, (short)0, c,
                                                false, false);
    }
    wp += (size_t)UNROLL * 4 * HEAD_OUT;
    hp += UNROLL * 8;
  }
  for (; it < cnt; ++it) {
    v2f a;
    a.x = hp[0] * asc;
    a.y = hp[2] * asc;
    v2f b;
    b.x = __builtin_nontemporal_load(wp);
    b.y = __builtin_nontemporal_load(wp + HEAD_OUT);
    c = __builtin_amdgcn_wmma_f32_16x16x4_f32(false, a, false, b, (short)0, c,
                                              false, false);
    wp += (size_t)4 * HEAD_OUT;
    hp += 8;
  }

  // D layout: lanes 0-15, c[0] = row M=0 (feat1), c[1] = row M=1 (feat2)
  if (lane < 16) {
    atomicAdd(&o1[col0 + lane], c[0]);
    atomicAdd(&o2[col0 + lane], c[1]);
  }
}

// ---------------- final dot over 1024 + sigmoid ---------------------------
__global__ void final_kernel(const float* __restrict__ o1, const float* __restrict__ o2,
                             const float* __restrict__ nb, float* out) {
  __shared__ float sdata[256];
  int t = threadIdx.x;
  float s = 0.f;
  for (int k = t; k < HEAD_OUT; k += 256)
    s += (o1[k] + nb[k]) * (o2[k] + nb[k]);
  sdata[t] = s;
  __syncthreads();
  for (int off = 128; off > 0; off >>= 1) {
    if (t < off) sdata[t] += sdata[t + off];
    __syncthreads();
  }
  if (t == 0) out[0] = 1.0f / (1.0f + expf(-sdata[0]));
}

extern "C" void kernel_launch(void* const* d_in, const int* in_sizes, int n_in,
                              void* d_out, int out_size, void* d_ws, size_t ws_size,
                              hipStream_t stream) {
  const float* feat1 = (const float*)d_in[0];
  const float* feat2 = (const float*)d_in[1];
  const float* norm  = (const float*)d_in[2];
  const int*   src   = (const int*)  d_in[3];
  const int*   dst   = (const int*)  d_in[4];
  const float* gin_w = (const float*)d_in[5];   // [3]
  const float* gin_b = (const float*)d_in[6];   // [3]
  const float* net_w = (const float*)d_in[7];   // [N, 1024] row-major
  const float* net_b = (const float*)d_in[8];   // [1024]
  float* out = (float*)d_out;

  const int N = in_sizes[0];   // 100000
  const int E = in_sizes[2];   // 3200000

  // workspace carve (float2 arrays first, 8B-aligned): 7*N + 2*1024 floats
  v2f*   xv   = (v2f*)d_ws;            // [N] float2
  v2f*   aggv = xv   + N;              // [N] float2
  v2f*   sumv = aggv + N;              // [N] float2
  float* deg  = (float*)(sumv + N);    // [N]
  float* o1   = deg + N;               // [1024]
  float* o2   = o1  + HEAD_OUT;        // [1024]

  const int TB = 256;
  int nBlocksN = (N + TB - 1) / TB;
  int nBlocksE = (E + TB - 1) / TB;

  init_kernel<<<nBlocksN, TB, 0, stream>>>(feat1, feat2, deg, xv, aggv, sumv, o1, o2, N);
  deg_kernel<<<nBlocksE, TB, 0, stream>>>(dst, deg, E);

  for (int l = 0; l < N_HID; ++l) {
    edge_kernel<<<nBlocksE, TB, 0, stream>>>(src, dst, norm, xv, (float*)aggv, E);
    node_kernel<<<nBlocksN, TB, 0, stream>>>(deg, xv, aggv, sumv, gin_w, gin_b, l, N);
  }

  head_wmma_kernel<<<dim3(HEAD_TILES, HEAD_GY), TB, 0, stream>>>((const float*)sumv,
                                                                 net_w, o1, o2, N);
  final_kernel<<<1, TB, 0, stream>>>(o1, o2, net_b, out);
}